// SuperPixelMeanEmbed_38620345925873
// MI455X (gfx1250) — compile-verified
//
#include <hip/hip_runtime.h>

typedef __attribute__((ext_vector_type(2))) float v2f;
typedef __attribute__((ext_vector_type(8))) float v8f;

// Problem constants (from reference setup_inputs)
constexpr int Bn = 16;           // batch
constexpr int Ci = 3;            // input channels
constexpr int Hh = 224, Ww = 224;
constexpr int S  = 196;          // superpixels per image
constexpr int Co = 56;           // output channels (224/4)
constexpr int HW = Hh * Ww;      // 50176 pixels per image
constexpr int ACC_FLOATS = Bn * S * 4;  // [sx0,sx1,sx2,count] per (img,seg) = 12544 floats

// ---------------------------------------------------------------------------
// Kernel 0: zero the accumulator workspace (must run every launch: harness
// poisons d_ws once and we atomically accumulate into it).
// ---------------------------------------------------------------------------
__global__ void spme_zero_ws(float* __restrict__ ws) {
    int i = blockIdx.x * blockDim.x + threadIdx.x;
    if (i < ACC_FLOATS) ws[i] = 0.0f;
}

// ---------------------------------------------------------------------------
// Kernel 1: segment reduction. Each block owns one image-slice; per-segment
// partial sums [sx0,sx1,sx2,count] privatized in LDS (196*4 floats = 3.1 KB,
// ds_add_f32), then flushed with one global atomic pass per block.
// Memory-bound: reads X (9.6 MB) + seg_ids (3.2 MB), fully coalesced.
// ---------------------------------------------------------------------------
__global__ void spme_accum(const float* __restrict__ X,
                           const int*   __restrict__ seg,
                           float*       __restrict__ acc) {
    __shared__ float lacc[S * 4];
    const int img = blockIdx.y;
    const int tid = threadIdx.x;

    for (int i = tid; i < S * 4; i += blockDim.x) lacc[i] = 0.0f;
    __syncthreads();

    const float* x  = X   + (size_t)img * Ci * HW;
    const int*   sg = seg + (size_t)img * HW;
    const int chunk = HW / gridDim.x;        // 50176/16 = 3136 pixels per block
    const int p0 = blockIdx.x * chunk;
    const int p1 = p0 + chunk;

    for (int p = p0 + tid; p < p1; p += blockDim.x) {
        const int s4 = sg[p] * 4;
        atomicAdd(&lacc[s4 + 0], x[p]);
        atomicAdd(&lacc[s4 + 1], x[p + HW]);
        atomicAdd(&lacc[s4 + 2], x[p + 2 * HW]);
        atomicAdd(&lacc[s4 + 3], 1.0f);
    }
    __syncthreads();

    float* g = acc + (size_t)img * S * 4;
    for (int i = tid; i < S * 4; i += blockDim.x)
        atomicAdd(&g[i], lacc[i]);
}

// ---------------------------------------------------------------------------
// Kernel 2: finalize with V_WMMA_F32_16X16X4_F32.
//   D(16x16) = A(16 segments x K=4) * B(K=4 x 16 channels)
//   A row m = [sx0, sx1, sx2, count] ; B col o = [W[o,0], W[o,1], W[o,2], b[o]]
//   => D[m][o] = W·Σx + count·b ; then scale row by 1/max(count,1).
// One wave32 per M-tile (3136 segs = 196 exact tiles), 4 N-tiles (Co=56<64).
// ISA VGPR layouts (CDNA5 ISA):
//   A 16x4 f32 : lanes 0-15 -> M=lane, v0=K0,v1=K1 ; lanes 16-31 -> v0=K2,v1=K3
//   B 4x16 f32 : N=lane&15  ; lanes 0-15 hold K={0,1}, lanes 16-31 K={2,3}
//   D 16x16 f32: vgpr r, lanes 0-15 -> M=r ; lanes 16-31 -> M=r+8 ; N=lane&15
// ---------------------------------------------------------------------------
__global__ void spme_finalize_wmma(const float* __restrict__ acc,
                                   const float* __restrict__ Wmat,
                                   const float* __restrict__ bias,
                                   float*       __restrict__ out) {
    const int mtile = blockIdx.x;    // 0..195
    const int lane  = threadIdx.x;   // 0..31 (wave32, full EXEC)
    const int n     = lane & 15;
    const int khalf = lane >> 4;     // 0 | 1
    const int klo   = khalf * 2;     // 0 | 2

    // A operand: this lane's segment row, K-pair {klo, klo+1}
    const int segA = mtile * 16 + n;
    v2f A;
    A.x = acc[segA * 4 + klo];
    A.y = acc[segA * 4 + klo + 1];

    // Per-row 1/max(count,1) for the 8 D rows this lane holds (M = r + 8*khalf)
    float inv[8];
#pragma unroll
    for (int r = 0; r < 8; ++r) {
        const float cnt = acc[(mtile * 16 + r + khalf * 8) * 4 + 3];
        inv[r] = 1.0f / fmaxf(cnt, 1.0f);
    }

#pragma unroll
    for (int ntile = 0; ntile < 4; ++ntile) {
        const int o  = ntile * 16 + n;          // output channel of this lane's B column
        const int oc = (o < Co) ? o : (Co - 1); // clamp for safe loads
        // B rows {klo, klo+1}: k in {0,2} is always a W row; k=1 -> W, k=3 -> bias
        float b0 = Wmat[oc * Ci + klo];
        float b1 = (klo + 1 < Ci) ? Wmat[oc * Ci + 1] : bias[oc];
        if (o >= Co) { b0 = 0.0f; b1 = 0.0f; }  // pad columns 56..63 with zeros
        v2f Bv;
        Bv.x = b0;
        Bv.y = b1;

        v8f C = {};
        // (neg_a, A, neg_b, B, c_mod, C, reuse_a, reuse_b)
        v8f D = __builtin_amdgcn_wmma_f32_16x16x4_f32(
            false, A, false, Bv, (short)0, C, false, false);

        if (o < Co) {
#pragma unroll
            for (int r = 0; r < 8; ++r) {
                const int g = mtile * 16 + r + khalf * 8;   // flat (img*S + seg)
                out[(size_t)g * Co + o] = D[r] * inv[r];
            }
        }
    }
}

// ---------------------------------------------------------------------------
extern "C" void kernel_launch(void* const* d_in, const int* in_sizes, int n_in,
                              void* d_out, int out_size, void* d_ws, size_t ws_size,
                              hipStream_t stream) {
    const float* X    = (const float*)d_in[0];  // (16,3,224,224) f32
    const float* Wmat = (const float*)d_in[1];  // (56,3) f32
    const float* bias = (const float*)d_in[2];  // (56,) f32
    const int*   seg  = (const int*)d_in[3];    // (16,224,224) i32
    float* acc = (float*)d_ws;                  // 12544 f32 = 50 KB
    float* out = (float*)d_out;                 // (16,196,56) f32

    spme_zero_ws<<<(ACC_FLOATS + 255) / 256, 256, 0, stream>>>(acc);
    spme_accum<<<dim3(16, Bn), 256, 0, stream>>>(X, seg, acc);
    spme_finalize_wmma<<<196, 32, 0, stream>>>(acc, Wmat, bias, out);
}